// GAT_IMPUTER_78606491452019
// MI455X (gfx1250) — compile-verified
//
#include <hip/hip_runtime.h>
#include <hip/hip_bf16.h>
#include <math.h>

// Problem constants (match reference)
#define NNODES 50000
#define DIN    256
#define NH     4
#define NC     16
#define HC     64          // NH*NC
#define NEDGE  1600000
#define NEG_SLOPE 0.2f
#define MT     5           // M-tiles per wave (3125 m-tiles = 5^5, divisible by 5)

typedef __attribute__((ext_vector_type(2))) float v2f;
typedef __attribute__((ext_vector_type(8))) float v8f;

__device__ __forceinline__ float sigmoidf_(float v) {
    return 1.0f / (1.0f + __expf(-v));
}

// Float atomic max via int punning (works for all signs; init buffer to -inf).
__device__ __forceinline__ void atomicMaxF(float* addr, float value) {
    if (value >= 0.0f)
        atomicMax((int*)addr, __float_as_int(value));
    else
        atomicMin((unsigned int*)addr, __float_as_uint(value));
}

// -----------------------------------------------------------------------------
// WMMA fp32 GEMM: C[M,Ncols] = A[M,K] @ W[K,Ncols] + bias   (EPI=1 -> sigmoid)
// Each wave computes MT(=5) vertically-stacked 16x16 tiles sharing one N-tile,
// so the B (weight) fragment is loaded once per K-step and reused across 5
// V_WMMA_F32_16X16X4_F32 ops with independent accumulators.
// M multiple of 16*MT, K multiple of 4, Ncols multiple of 16. All guards are
// wave-uniform so EXEC is all-ones for every WMMA.
// -----------------------------------------------------------------------------
template<int EPI>
__global__ void wmma_gemm_kernel(const float* __restrict__ A,
                                 const float* __restrict__ W,
                                 const float* __restrict__ bias,
                                 float* __restrict__ Cout,
                                 int M, int K, int Ncols) {
    const int lane  = threadIdx.x & 31;
    const int wave  = threadIdx.x >> 5;
    const int tilesN  = Ncols >> 4;
    const int mGroups = M / (16 * MT);
    const int tile = blockIdx.x * (blockDim.x >> 5) + wave;
    const int totalTiles = mGroups * tilesN;
    if (tile >= totalTiles) return;                 // wave-uniform
    const int tg = tile / tilesN;                   // m-group index
    const int tn = tile - tg * tilesN;
    const int m0 = tg * (16 * MT);
    const int n0 = tn << 4;

    const int row  = lane & 15;       // A row / B col / D col within tile
    const int half = lane >> 4;       // selects K pair {0,1} vs {2,3}

    // A fragment sources: lane holds A[m0 + j*16 + row][k + 2*half + {0,1}]
    const float2* arow[MT];
    #pragma unroll
    for (int j = 0; j < MT; ++j)
        arow[j] = (const float2*)(A + (size_t)(m0 + j * 16 + row) * K + 2 * half);
    // B fragment source: lane holds W[k + 2*half + {0,1}][n0+row]
    const float* wcol = W + n0 + row;

    v8f acc[MT];
    #pragma unroll
    for (int j = 0; j < MT; ++j)
        acc[j] = (v8f){0.f, 0.f, 0.f, 0.f, 0.f, 0.f, 0.f, 0.f};

    for (int k = 0; k < K; k += 4) {
        v2f b;
        b.x = wcol[(size_t)(k + 2 * half)     * Ncols];
        b.y = wcol[(size_t)(k + 2 * half + 1) * Ncols];
        #pragma unroll
        for (int j = 0; j < MT; ++j) {
            float2 av = arow[j][k >> 1];
            v2f a; a.x = av.x; a.y = av.y;
            acc[j] = __builtin_amdgcn_wmma_f32_16x16x4_f32(
                /*neg_a=*/false, a, /*neg_b=*/false, b,
                /*c_mod=*/(short)0, acc[j], /*reuse_a=*/false, /*reuse_b=*/false);
        }
    }

    const float bv = bias[n0 + row];
    #pragma unroll
    for (int j = 0; j < MT; ++j) {
        #pragma unroll
        for (int r = 0; r < 8; ++r) {
            int m = m0 + j * 16 + r + half * 8; // D layout: VGPR r -> M=r / M=r+8
            float v = acc[j][r] + bv;
            if (EPI == 1) v = sigmoidf_(v);
            Cout[(size_t)m * Ncols + n0 + row] = v;
        }
    }
}

// -----------------------------------------------------------------------------
// Per-layer scratch init: accum=0 (N*HC), maxv=-inf, denom=0 (N*NH)
// -----------------------------------------------------------------------------
__global__ void init_buffers_kernel(float* __restrict__ accum,
                                    float* __restrict__ maxv,
                                    float* __restrict__ denom,
                                    int n64, int n4) {
    int i = blockIdx.x * blockDim.x + threadIdx.x;
    if (i < n64) accum[i] = 0.0f;
    if (i < n4) { maxv[i] = -__builtin_inff(); denom[i] = 0.0f; }
}

// -----------------------------------------------------------------------------
// Pass 1: logits[e,h] = att[h] . leaky_relu(xl[src[e],h,:] + xr[dst[e],h,:])
//         and scatter-max into maxv[dst,h]
// One thread per (edge, head).
// -----------------------------------------------------------------------------
__global__ void edge_logits_kernel(const long long* __restrict__ src,
                                   const long long* __restrict__ dst,
                                   const float* __restrict__ xl,
                                   const float* __restrict__ xr,
                                   const float* __restrict__ att,
                                   float* __restrict__ elog,
                                   float* __restrict__ maxv) {
    int t = blockIdx.x * blockDim.x + threadIdx.x;
    if (t >= NEDGE * NH) return;
    int h = t & (NH - 1);
    int e = t >> 2;
    int s = (int)src[e];
    int d = (int)dst[e];
    const float4* pl = (const float4*)(xl + (size_t)s * HC + h * NC);
    const float4* pr = (const float4*)(xr + (size_t)d * HC + h * NC);
    const float4* pa = (const float4*)(att + h * NC);
    float acc = 0.0f;
    #pragma unroll
    for (int i = 0; i < 4; ++i) {
        float4 a = pl[i], b = pr[i], w = pa[i];
        float v;
        v = a.x + b.x; v = v > 0.0f ? v : NEG_SLOPE * v; acc += w.x * v;
        v = a.y + b.y; v = v > 0.0f ? v : NEG_SLOPE * v; acc += w.y * v;
        v = a.z + b.z; v = v > 0.0f ? v : NEG_SLOPE * v; acc += w.z * v;
        v = a.w + b.w; v = v > 0.0f ? v : NEG_SLOPE * v; acc += w.w * v;
    }
    elog[t] = acc;
    atomicMaxF(&maxv[(size_t)d * NH + h], acc);
}

// -----------------------------------------------------------------------------
// Pass 2: ex = exp(logit - max[dst,h]); elog <- ex; denom[dst,h] += ex
// -----------------------------------------------------------------------------
__global__ void edge_exp_kernel(const long long* __restrict__ dst,
                                float* __restrict__ elog,
                                const float* __restrict__ maxv,
                                float* __restrict__ denom) {
    int t = blockIdx.x * blockDim.x + threadIdx.x;
    if (t >= NEDGE * NH) return;
    int h = t & (NH - 1);
    int e = t >> 2;
    int d = (int)dst[e];
    float ex = __expf(elog[t] - maxv[(size_t)d * NH + h]);
    elog[t] = ex;
    unsafeAtomicAdd(&denom[(size_t)d * NH + h], ex);
}

// -----------------------------------------------------------------------------
// Pass 3: alpha = ex / denom[dst,h]; accum[dst,h,:] += alpha * xl[src,h,:]
// -----------------------------------------------------------------------------
__global__ void edge_scatter_kernel(const long long* __restrict__ src,
                                    const long long* __restrict__ dst,
                                    const float* __restrict__ elog,
                                    const float* __restrict__ denom,
                                    const float* __restrict__ xl,
                                    float* __restrict__ accum) {
    int t = blockIdx.x * blockDim.x + threadIdx.x;
    if (t >= NEDGE * NH) return;
    int h = t & (NH - 1);
    int e = t >> 2;
    int s = (int)src[e];
    int d = (int)dst[e];
    float alpha = elog[t] / denom[(size_t)d * NH + h];
    const float4* pl = (const float4*)(xl + (size_t)s * HC + h * NC);
    float* po = accum + (size_t)d * HC + h * NC;
    #pragma unroll
    for (int i = 0; i < 4; ++i) {
        float4 v = pl[i];
        unsafeAtomicAdd(po + 4 * i + 0, alpha * v.x);
        unsafeAtomicAdd(po + 4 * i + 1, alpha * v.y);
        unsafeAtomicAdd(po + 4 * i + 2, alpha * v.z);
        unsafeAtomicAdd(po + 4 * i + 3, alpha * v.w);
    }
}

// -----------------------------------------------------------------------------
// h[n,:] = relu(accum[n,:] + bias[:])
// -----------------------------------------------------------------------------
__global__ void bias_relu_kernel(const float* __restrict__ accum,
                                 const float* __restrict__ bias,
                                 float* __restrict__ hout) {
    int i = blockIdx.x * blockDim.x + threadIdx.x;
    if (i >= NNODES * HC) return;
    float v = accum[i] + bias[i & (HC - 1)];
    hout[i] = v > 0.0f ? v : 0.0f;
}

// -----------------------------------------------------------------------------
extern "C" void kernel_launch(void* const* d_in, const int* in_sizes, int n_in,
                              void* d_out, int out_size, void* d_ws, size_t ws_size,
                              hipStream_t stream) {
    (void)in_sizes; (void)n_in; (void)out_size; (void)ws_size;

    const float*     x    = (const float*)d_in[0];
    const long long* ei   = (const long long*)d_in[1];
    const long long* src  = ei;
    const long long* dst  = ei + NEDGE;
    const float* Wl1  = (const float*)d_in[2];
    const float* bl1  = (const float*)d_in[3];
    const float* Wr1  = (const float*)d_in[4];
    const float* br1  = (const float*)d_in[5];
    const float* att1 = (const float*)d_in[6];
    const float* bias1= (const float*)d_in[7];
    const float* Wl2  = (const float*)d_in[8];
    const float* bl2  = (const float*)d_in[9];
    const float* Wr2  = (const float*)d_in[10];
    const float* br2  = (const float*)d_in[11];
    const float* att2 = (const float*)d_in[12];
    const float* bias2= (const float*)d_in[13];
    const float* Wo   = (const float*)d_in[14];
    const float* bo   = (const float*)d_in[15];
    float* out = (float*)d_out;

    // Workspace layout (floats), ~91 MB total
    float* ws = (float*)d_ws;
    size_t o = 0;
    float* xl    = ws + o; o += (size_t)NNODES * HC;
    float* xr    = ws + o; o += (size_t)NNODES * HC;
    float* accum = ws + o; o += (size_t)NNODES * HC;
    float* h1    = ws + o; o += (size_t)NNODES * HC;
    float* h2    = ws + o; o += (size_t)NNODES * HC;
    float* maxv  = ws + o; o += (size_t)NNODES * NH;
    float* denom = ws + o; o += (size_t)NNODES * NH;
    float* elog  = ws + o; o += (size_t)NEDGE  * NH;

    const int TB = 256;
    const int wavesPerBlock = TB / 32;
    const int mGroups = NNODES / (16 * MT);                 // 625
    const int tilesLayer = mGroups * (HC / 16);             // 2500
    const int gemmBlocksLayer = (tilesLayer + wavesPerBlock - 1) / wavesPerBlock;
    const int nhcBlocks = (NNODES * HC + TB - 1) / TB;
    const int ehBlocks  = (NEDGE * NH + TB - 1) / TB;

    // ---------------- Layer 1 ----------------
    wmma_gemm_kernel<0><<<gemmBlocksLayer, TB, 0, stream>>>(x, Wl1, bl1, xl, NNODES, DIN, HC);
    wmma_gemm_kernel<0><<<gemmBlocksLayer, TB, 0, stream>>>(x, Wr1, br1, xr, NNODES, DIN, HC);
    init_buffers_kernel<<<nhcBlocks, TB, 0, stream>>>(accum, maxv, denom, NNODES * HC, NNODES * NH);
    edge_logits_kernel<<<ehBlocks, TB, 0, stream>>>(src, dst, xl, xr, att1, elog, maxv);
    edge_exp_kernel<<<ehBlocks, TB, 0, stream>>>(dst, elog, maxv, denom);
    edge_scatter_kernel<<<ehBlocks, TB, 0, stream>>>(src, dst, elog, denom, xl, accum);
    bias_relu_kernel<<<nhcBlocks, TB, 0, stream>>>(accum, bias1, h1);

    // ---------------- Layer 2 ----------------
    wmma_gemm_kernel<0><<<gemmBlocksLayer, TB, 0, stream>>>(h1, Wl2, bl2, xl, NNODES, HC, HC);
    wmma_gemm_kernel<0><<<gemmBlocksLayer, TB, 0, stream>>>(h1, Wr2, br2, xr, NNODES, HC, HC);
    init_buffers_kernel<<<nhcBlocks, TB, 0, stream>>>(accum, maxv, denom, NNODES * HC, NNODES * NH);
    edge_logits_kernel<<<ehBlocks, TB, 0, stream>>>(src, dst, xl, xr, att2, elog, maxv);
    edge_exp_kernel<<<ehBlocks, TB, 0, stream>>>(dst, elog, maxv, denom);
    edge_scatter_kernel<<<ehBlocks, TB, 0, stream>>>(src, dst, elog, denom, xl, accum);
    bias_relu_kernel<<<nhcBlocks, TB, 0, stream>>>(accum, bias2, h2);

    // ---------------- Output head: sigmoid(h2 @ Wo + bo) ----------------
    const int tilesOut = mGroups * (DIN / 16);              // 10000
    const int gemmBlocksOut = (tilesOut + wavesPerBlock - 1) / wavesPerBlock;
    wmma_gemm_kernel<1><<<gemmBlocksOut, TB, 0, stream>>>(h2, Wo, bo, out, NNODES, HC, DIN);
}